// NeusRender_83683142795556
// MI455X (gfx1250) — compile-verified
//
#include <hip/hip_runtime.h>
#include <hip/hip_bf16.h>

// NeuS renderer for MI455X (gfx1250). Pure HBM-bandwidth-bound (~290MB @ 23.3TB/s
// => ~12.5us floor; ~0.1 flop/byte so WMMA is inapplicable). Strategy:
//   - one wave32 per ray, lane l owns samples [4l,4l+4): all traffic is
//     coalesced b128 per lane
//   - cumprod -> wave exclusive prefix product via __shfl_up scan
//   - color+grad (24B/sample, bulk of traffic) staged via CDNA5 async
//     global->LDS DMA (GLOBAL_LOAD_ASYNC_TO_LDS_B128 / s_wait_asynccnt),
//     double-buffered across 4 rays per wave to overlap DMA with compute
//   - nontemporal store for the 32MB weight output (single-pass data)

typedef float v4f __attribute__((ext_vector_type(4)));
typedef int   vsi4 __attribute__((vector_size(16)));   // b128 payload type for async builtin

#define N_SAMP 128
#define WPB    8            // waves per block (256 threads)
#define TPB    (WPB * 32)
#define RPW    4            // rays per wave (pipeline depth over double buffer)

#if defined(__gfx1250__) && \
    __has_builtin(__builtin_amdgcn_global_load_async_to_lds_b128) && \
    __has_builtin(__builtin_amdgcn_s_wait_asynccnt)
#define USE_ASYNC 1
#else
#define USE_ASYNC 0
#endif

#if USE_ASYNC
#define ASYNC_WAIT(n) __builtin_amdgcn_s_wait_asynccnt(n)
typedef __attribute__((address_space(1))) vsi4* gptr_t;
typedef __attribute__((address_space(3))) vsi4* lptr_t;
#else
#define ASYNC_WAIT(n) ((void)0)
#endif

__device__ __forceinline__ float sigmoid_f(float x) {
    return 1.0f / (1.0f + __expf(-x));
}

// copy 16 bytes global -> LDS at byte offset OFF (same offset both sides,
// matching the ISA's INST_OFFSET semantics for async loads)
template <int OFF>
__device__ __forceinline__ void cp16(const float* g, float* l) {
#if USE_ASYNC
    __builtin_amdgcn_global_load_async_to_lds_b128(
        (gptr_t)(g), (lptr_t)(l), OFF, 0);
#else
    *(v4f*)((char*)l + OFF) =
        __builtin_nontemporal_load((const v4f*)((const char*)g + OFF));
#endif
}

__global__ __launch_bounds__(TPB) void neus_render_kernel(
    const float* __restrict__ sdf,      // [N,128]
    const float* __restrict__ color,    // [N,128,3]
    const float* __restrict__ z_vals,   // [N,128]
    const float* __restrict__ grad,     // [N,128,3]
    const float* __restrict__ rays_d,   // [N,3]
    const float* __restrict__ s_inv,    // [1]
    const float* __restrict__ car_p,    // [1] (cos_anneal_ratio)
    float* __restrict__ out_pixel,      // [N,3]
    float* __restrict__ out_invdepth,   // [N]
    float* __restrict__ out_weight)     // [N,128]
{
    // per-wave double buffer: 384 color floats + 384 grad floats per ray
    __shared__ float smem[WPB * 2 * 768];   // 48 KB / block

    const int lane = threadIdx.x & 31;
    const int wib  = threadIdx.x >> 5;
    const int gw   = blockIdx.x * WPB + wib;
    const int ray0 = gw * RPW;

    const float sv  = s_inv[0];
    const float car = car_p[0];

    float* const b0 = &smem[wib * 2 * 768];
    float* const b1 = b0 + 768;

    auto issue = [&](int ray, float* buf) {
        const float* cg = color + (size_t)ray * 384 + lane * 12;
        const float* gg = grad  + (size_t)ray * 384 + lane * 12;
        float* lc = buf + lane * 12;
        float* lg = buf + 384 + lane * 12;
        cp16<0>(cg, lc); cp16<16>(cg, lc); cp16<32>(cg, lc);
        cp16<0>(gg, lg); cp16<16>(gg, lg); cp16<32>(gg, lg);
    };

    auto process = [&](int ray, const float* buf) {
        const float* lc = buf + lane * 12;
        const float* lg = buf + 384 + lane * 12;

        v4f sd = __builtin_nontemporal_load(
            (const v4f*)(sdf + (size_t)ray * N_SAMP) + lane);
        v4f zz = __builtin_nontemporal_load(
            (const v4f*)(z_vals + (size_t)ray * N_SAMP) + lane);
        float rd0 = rays_d[ray * 3 + 0];
        float rd1 = rays_d[ray * 3 + 1];
        float rd2 = rays_d[ray * 3 + 2];
        float z_next = __shfl_down(zz.x, 1, 32);   // z[4l+4] (lane31: unused)

        v4f c0 = *(const v4f*)(lc + 0);
        v4f c1 = *(const v4f*)(lc + 4);
        v4f c2 = *(const v4f*)(lc + 8);
        v4f g0 = *(const v4f*)(lg + 0);
        v4f g1 = *(const v4f*)(lg + 4);
        v4f g2 = *(const v4f*)(lg + 8);

        float sdfv[4] = {sd.x, sd.y, sd.z, sd.w};
        float zl[5]   = {zz.x, zz.y, zz.z, zz.w, z_next};
        float gx[4]   = {g0.x, g0.w, g1.z, g2.y};
        float gy[4]   = {g0.y, g1.x, g1.w, g2.z};
        float gzv[4]  = {g0.z, g1.y, g2.x, g2.w};
        float cx[4]   = {c0.x, c0.w, c1.z, c2.y};
        float cy[4]   = {c0.y, c1.x, c1.w, c2.z};
        float cz[4]   = {c0.z, c1.y, c2.x, c2.w};

        float alpha[4];
#pragma unroll
        for (int j = 0; j < 4; ++j) {
            float tc = fmaf(rd0, gx[j], fmaf(rd1, gy[j], rd2 * gzv[j]));
            float r1 = fmaxf(fmaf(-tc, 0.5f, 0.5f), 0.0f);
            float r2 = fmaxf(-tc, 0.0f);
            float ic = -(r1 * (1.0f - car) + r2 * car);
            float dist = zl[j + 1] - zl[j];
            float e  = ic * dist * 0.5f;
            float pc = sigmoid_f((sdfv[j] - e) * sv);
            float nc = sigmoid_f((sdfv[j] + e) * sv);
            float a  = (pc - nc + 1e-5f) / (pc + 1e-5f);
            alpha[j] = fminf(fmaxf(a, 0.0f), 1.0f);
        }
        if (lane == 31) alpha[3] = 0.0f;   // appended zero column (sample 127)

        float om0 = 1.0f - alpha[0], om1 = 1.0f - alpha[1];
        float om2 = 1.0f - alpha[2], om3 = 1.0f - alpha[3];

        // exclusive prefix product of (1-alpha) across the 128 samples:
        // per-lane product, wave scan, local expansion
        float t = om0 * om1 * om2 * om3;
#pragma unroll
        for (int off = 1; off < 32; off <<= 1) {
            float u = __shfl_up(t, off, 32);
            if (lane >= off) t *= u;
        }
        float excl = __shfl_up(t, 1, 32);
        if (lane == 0) excl = 1.0f;

        float T0 = excl;
        float T1 = T0 * om0;
        float T2 = T1 * om1;
        float T3 = T2 * om2;
        float w0 = (lane == 0) ? 0.0f : alpha[0] * T0; // T_full[0] = 0
        float w1 = alpha[1] * T1;
        float w2 = alpha[2] * T2;
        float w3 = alpha[3] * T3;

        v4f wv = {w0, w1, w2, w3};
        __builtin_nontemporal_store(
            wv, (v4f*)(out_weight + (size_t)ray * N_SAMP) + lane);

        float px = w0 * cx[0] + w1 * cx[1] + w2 * cx[2] + w3 * cx[3];
        float py = w0 * cy[0] + w1 * cy[1] + w2 * cy[2] + w3 * cy[3];
        float pz = w0 * cz[0] + w1 * cz[1] + w2 * cz[2] + w3 * cz[3];
        float id = __fdividef(w0, zl[0]) + __fdividef(w1, zl[1]) +
                   __fdividef(w2, zl[2]) + __fdividef(w3, zl[3]);
#pragma unroll
        for (int off = 16; off >= 1; off >>= 1) {
            px += __shfl_xor(px, off, 32);
            py += __shfl_xor(py, off, 32);
            pz += __shfl_xor(pz, off, 32);
            id += __shfl_xor(id, off, 32);
        }
        if (lane == 0) {
            out_pixel[ray * 3 + 0] = px;
            out_pixel[ray * 3 + 1] = py;
            out_pixel[ray * 3 + 2] = pz;
            out_invdepth[ray] = id;
        }
    };

    // software pipeline: DMA ray i+1 while computing ray i
    issue(ray0, b0);
#pragma unroll
    for (int i = 0; i < RPW; ++i) {
        float* cur = (i & 1) ? b1 : b0;
        float* nxt = (i & 1) ? b0 : b1;
        if (i + 1 < RPW) {
            issue(ray0 + i + 1, nxt);
            ASYNC_WAIT(6);          // 6 newly-issued may remain outstanding
        } else {
            ASYNC_WAIT(0);
        }
        __asm__ volatile("" ::: "memory");  // keep LDS reads below the wait
        process(ray0 + i, cur);
    }
}

extern "C" void kernel_launch(void* const* d_in, const int* in_sizes, int n_in,
                              void* d_out, int out_size, void* d_ws, size_t ws_size,
                              hipStream_t stream) {
    const float* sdf    = (const float*)d_in[0];
    const float* color  = (const float*)d_in[1];
    const float* z_vals = (const float*)d_in[2];
    const float* grad   = (const float*)d_in[3];
    const float* rays_d = (const float*)d_in[4];
    const float* s_inv  = (const float*)d_in[5];
    const float* car    = (const float*)d_in[6];

    const int n_rays = in_sizes[0] / N_SAMP;       // 65536
    float* out       = (float*)d_out;
    float* pixel     = out;                         // [N,3]
    float* invdepth  = out + (size_t)n_rays * 3;    // [N]
    float* weight    = out + (size_t)n_rays * 4;    // [N,128]

    const int blocks = n_rays / (WPB * RPW);        // 2048
    neus_render_kernel<<<blocks, TPB, 0, stream>>>(
        sdf, color, z_vals, grad, rays_d, s_inv, car,
        pixel, invdepth, weight);
}